// Overall_model_spartun_47931835023578
// MI455X (gfx1250) — compile-verified
//
#include <hip/hip_runtime.h>
#include <hip/hip_bf16.h>

typedef float v8f __attribute__((ext_vector_type(8)));
typedef float v2f __attribute__((ext_vector_type(2)));

#define Bsz 4
#define Nn  32
#define Ee  128
#define Dd  768
#define D3  (3 * Dd)

#define WMMA_F32(a, b, c) \
  __builtin_amdgcn_wmma_f32_16x16x4_f32(false, (a), false, (b), (short)0, (c), false, false)

// ---------------------------------------------------------------------------
// Kernel 1: build mlp_in rows [h_src | edge | h_dst] and alpha[b,e] = h_dst . target
// one block (256 thr) per edge; B*E = 512 blocks
// ---------------------------------------------------------------------------
__global__ __launch_bounds__(256) void build_edge_inputs(
    const float* __restrict__ node, const float* __restrict__ edgef,
    const int* __restrict__ eidx, const float* __restrict__ target,
    float* __restrict__ mlp_in, float* __restrict__ alpha) {
  const int r = blockIdx.x;          // b*E + e
  const int b = r / Ee;
  const int e = r - b * Ee;
  const int tid = threadIdx.x;

  const int s = eidx[b * 2 * Ee + e];        // src
  const int d = eidx[b * 2 * Ee + Ee + e];   // dst

  const float* ns = node + ((size_t)b * Nn + s) * Dd;
  const float* nd = node + ((size_t)b * Nn + d) * Dd;
  const float* ef = edgef + (size_t)r * Dd;
  const float* tg = target + (size_t)b * Dd;
  float* row = mlp_in + (size_t)r * D3;

  float part = 0.0f;
  for (int c = tid; c < Dd; c += 256) {
    float hd = nd[c];
    row[c]          = ns[c];
    row[Dd + c]     = ef[c];
    row[2 * Dd + c] = hd;
    part += hd * tg[c];
  }
  __shared__ float red[256];
  red[tid] = part;
  __syncthreads();
  for (int off = 128; off > 0; off >>= 1) {
    if (tid < off) red[tid] += red[tid + off];
    __syncthreads();
  }
  if (tid == 0) alpha[r] = red[0];
}

// ---------------------------------------------------------------------------
// WMMA f32 GEMM:  C[M,N] = A[M,K] @ W[K,N] + bias[N]
// Register-blocked: each wave computes a 64x32 macro-tile (4 M-tiles x 2
// N-tiles, 8 v8f accumulators). Per K-step of 4: 4 A-frag b64 loads + 4 B b32
// loads feed 8 WMMAs. Tile IDs flattened over waves; block = 256 (8 waves).
// Requires M % 64 == 0, N % 32 == 0, K % 4 == 0, (M/64)*(N/32) % 8 == 0.
// A frag (ISA 7.12.2, 32-bit A 16x4): lane l(0..15)->M=l ; VGPR0/1 hold
//   K = k0 + 2*(lane>>4), +1.  C/D: lane->N, VGPR v -> M = v + 8*(lane>>4).
// ---------------------------------------------------------------------------
__global__ __launch_bounds__(256) void gemm_wmma_f32(
    const float* __restrict__ A, const float* __restrict__ W,
    const float* __restrict__ bias, float* __restrict__ C,
    int M, int K, int Ncols) {
  const int lane = threadIdx.x & 31;
  const int wave = threadIdx.x >> 5;
  const int tilesN = Ncols >> 5;                     // N / 32
  const int id = blockIdx.x * 8 + wave;              // macro-tile id
  const int rowBase = (id / tilesN) << 6;            // * 64
  const int colBase = (id % tilesN) << 5;            // * 32
  const int half = lane >> 4;                        // 0 | 1
  const int l = lane & 15;

  // per-lane advancing pointers (constant-stride -> immediate-offset loads)
  const float* pA0 = A + (size_t)(rowBase +      l) * K + half * 2;
  const float* pA1 = A + (size_t)(rowBase + 16 + l) * K + half * 2;
  const float* pA2 = A + (size_t)(rowBase + 32 + l) * K + half * 2;
  const float* pA3 = A + (size_t)(rowBase + 48 + l) * K + half * 2;
  const float* pW0 = W + (size_t)(half * 2) * Ncols + colBase + l;
  const float* pW1 = pW0 + 16;

  v8f acc0 = {}, acc1 = {}, acc2 = {}, acc3 = {};
  v8f acc4 = {}, acc5 = {}, acc6 = {}, acc7 = {};

#pragma unroll 4
  for (int k0 = 0; k0 < K; k0 += 4) {
    v2f a0 = *(const v2f*)pA0;             // K = ka, ka+1 (8B aligned)
    v2f a1 = *(const v2f*)pA1;
    v2f a2 = *(const v2f*)pA2;
    v2f a3 = *(const v2f*)pA3;
    v2f b0; b0.x = pW0[0]; b0.y = pW0[Ncols];
    v2f b1; b1.x = pW1[0]; b1.y = pW1[Ncols];

    acc0 = WMMA_F32(a0, b0, acc0);
    acc1 = WMMA_F32(a0, b1, acc1);
    acc2 = WMMA_F32(a1, b0, acc2);
    acc3 = WMMA_F32(a1, b1, acc3);
    acc4 = WMMA_F32(a2, b0, acc4);
    acc5 = WMMA_F32(a2, b1, acc5);
    acc6 = WMMA_F32(a3, b0, acc6);
    acc7 = WMMA_F32(a3, b1, acc7);

    pA0 += 4; pA1 += 4; pA2 += 4; pA3 += 4;
    pW0 += (size_t)4 * Ncols;
    pW1 += (size_t)4 * Ncols;
  }

  const int col0 = colBase + l;
  const int col1 = col0 + 16;
  const float bb0 = bias[col0];
  const float bb1 = bias[col1];
  const int mrow = rowBase + half * 8;

#pragma unroll
  for (int v = 0; v < 8; ++v) {
    C[(size_t)(mrow      + v) * Ncols + col0] = acc0[v] + bb0;
    C[(size_t)(mrow      + v) * Ncols + col1] = acc1[v] + bb1;
    C[(size_t)(mrow + 16 + v) * Ncols + col0] = acc2[v] + bb0;
    C[(size_t)(mrow + 16 + v) * Ncols + col1] = acc3[v] + bb1;
    C[(size_t)(mrow + 32 + v) * Ncols + col0] = acc4[v] + bb0;
    C[(size_t)(mrow + 32 + v) * Ncols + col1] = acc5[v] + bb1;
    C[(size_t)(mrow + 48 + v) * Ncols + col0] = acc6[v] + bb0;
    C[(size_t)(mrow + 48 + v) * Ncols + col1] = acc7[v] + bb1;
  }
}

// ---------------------------------------------------------------------------
// Kernel 3: retrieved[b,n,:] = sum_{e: src==n} alpha[b,e]*filler[b,e,:]
// then LayerNorm, then build x2 row = [node | LN(retrieved) | target].
// One block (256 thr) per (b,n); branch on src is block-uniform -> no
// divergence, no atomics -> deterministic.
// ---------------------------------------------------------------------------
__global__ __launch_bounds__(256) void retrieve_ln_concat(
    const float* __restrict__ filler, const float* __restrict__ alpha,
    const int* __restrict__ eidx, const float* __restrict__ node,
    const float* __restrict__ target, const float* __restrict__ ln_g,
    const float* __restrict__ ln_b, float* __restrict__ x2) {
  const int row = blockIdx.x;        // b*N + n
  const int b = row / Nn;
  const int n = row - b * Nn;
  const int tid = threadIdx.x;

  const int* srcIdx = eidx + b * 2 * Ee;   // src row of edge_index
  float acc0 = 0.f, acc1 = 0.f, acc2 = 0.f;
  for (int e = 0; e < Ee; ++e) {
    if (srcIdx[e] == n) {                  // uniform across block
      const float a = alpha[b * Ee + e];
      const float* f = filler + (size_t)(b * Ee + e) * Dd;
      acc0 += a * f[tid];
      acc1 += a * f[tid + 256];
      acc2 += a * f[tid + 512];
    }
  }

  __shared__ float red[256];
  red[tid] = acc0 + acc1 + acc2;
  __syncthreads();
  for (int off = 128; off > 0; off >>= 1) {
    if (tid < off) red[tid] += red[tid + off];
    __syncthreads();
  }
  const float mu = red[0] * (1.0f / Dd);
  __syncthreads();
  const float d0 = acc0 - mu, d1 = acc1 - mu, d2 = acc2 - mu;
  red[tid] = d0 * d0 + d1 * d1 + d2 * d2;
  __syncthreads();
  for (int off = 128; off > 0; off >>= 1) {
    if (tid < off) red[tid] += red[tid + off];
    __syncthreads();
  }
  const float var = red[0] * (1.0f / Dd);
  const float inv = rsqrtf(var + 1e-5f);

  float* xr = x2 + (size_t)row * D3;
  const float* nf = node + (size_t)row * Dd;     // node[b,n,:] (row == b*N+n)
  const float* tg = target + (size_t)b * Dd;
  const float acc[3] = {acc0, acc1, acc2};
#pragma unroll
  for (int j = 0; j < 3; ++j) {
    const int c = tid + j * 256;
    const float ln = (acc[j] - mu) * inv * ln_g[c] + ln_b[c];
    xr[c]          = nf[c];
    xr[Dd + c]     = ln;
    xr[2 * Dd + c] = tg[c];
  }
}

// ---------------------------------------------------------------------------
extern "C" void kernel_launch(void* const* d_in, const int* in_sizes, int n_in,
                              void* d_out, int out_size, void* d_ws, size_t ws_size,
                              hipStream_t stream) {
  const float* node   = (const float*)d_in[0];
  const float* edgef  = (const float*)d_in[1];
  const int*   eidx   = (const int*)d_in[2];
  const float* target = (const float*)d_in[3];
  const float* Wf1 = (const float*)d_in[4];
  const float* bf1 = (const float*)d_in[5];
  const float* Wf2 = (const float*)d_in[6];
  const float* bf2 = (const float*)d_in[7];
  const float* Wf3 = (const float*)d_in[8];
  const float* bf3 = (const float*)d_in[9];
  const float* ln_g = (const float*)d_in[10];
  const float* ln_b = (const float*)d_in[11];
  const float* Wi1 = (const float*)d_in[12];
  const float* bi1 = (const float*)d_in[13];
  const float* Wi2 = (const float*)d_in[14];
  const float* bi2 = (const float*)d_in[15];
  const float* Wi3 = (const float*)d_in[16];
  const float* bi3 = (const float*)d_in[17];

  // workspace layout (floats)
  float* ws = (float*)d_ws;
  float* mlp_in = ws;                                  // 512 * 2304
  float* t1     = mlp_in + (size_t)Bsz * Ee * D3;      // 512 * 768
  float* t2     = t1 + (size_t)Bsz * Ee * Dd;          // 512 * 768
  float* alpha  = t2 + (size_t)Bsz * Ee * Dd;          // 512
  float* x2     = alpha + (size_t)Bsz * Ee;            // 128 * 2304
  float* u1     = x2 + (size_t)Bsz * Nn * D3;          // 128 * 768
  float* u2     = u1 + (size_t)Bsz * Nn * Dd;          // 128 * 768

  // 1) gather edge rows + alpha
  build_edge_inputs<<<Bsz * Ee, 256, 0, stream>>>(node, edgef, eidx, target,
                                                  mlp_in, alpha);

  // 2) filler = ((mlp_in@Wf1+bf1)@Wf2+bf2)@Wf3+bf3   (M=512)
  // macro-tiles: (512/64)*(768/32) = 192 waves -> 24 blocks
  gemm_wmma_f32<<<24, 256, 0, stream>>>(mlp_in, Wf1, bf1, t1, Bsz * Ee, D3, Dd);
  gemm_wmma_f32<<<24, 256, 0, stream>>>(t1, Wf2, bf2, t2, Bsz * Ee, Dd, Dd);
  gemm_wmma_f32<<<24, 256, 0, stream>>>(t2, Wf3, bf3, t1, Bsz * Ee, Dd, Dd);
  // t1 now holds filler (512 x 768)

  // 3) retrieved + layernorm + concat -> x2 (128 x 2304)
  retrieve_ln_concat<<<Bsz * Nn, 256, 0, stream>>>(t1, alpha, eidx, node,
                                                   target, ln_g, ln_b, x2);

  // 4) out = ((x2@Wi1+bi1)@Wi2+bi2)@Wi3+bi3   (M=128)
  // macro-tiles: (128/64)*(768/32) = 48 waves -> 6 blocks
  gemm_wmma_f32<<<6, 256, 0, stream>>>(x2, Wi1, bi1, u1, Bsz * Nn, D3, Dd);
  gemm_wmma_f32<<<6, 256, 0, stream>>>(u1, Wi2, bi2, u2, Bsz * Nn, Dd, Dd);
  gemm_wmma_f32<<<6, 256, 0, stream>>>(u2, Wi3, bi3, (float*)d_out,
                                       Bsz * Nn, Dd, Dd);
}